// AVSNet_438086664393
// MI455X (gfx1250) — compile-verified
//
#include <hip/hip_runtime.h>

// ---------------------------------------------------------------------------
// AVSNet adaptive voxel sampling for gfx1250 (MI455X, wave32).
//  - PI voxel-size loop: hash-set unique counting, device "done" flag (fixed
//    10 iteration unroll so it is graph-capturable).
//  - Centroids: hash-accumulate sums, bitonic-sort unique keys in LDS (64KB)
//    to reproduce the reference's sorted-segment ordering, binary-search rank.
//  - Nearest-point argmin: V_WMMA_F32_16X16X4_F32 computes 16x16 blocks of
//    xyz . cent^T (K=3 padded to 4). Points are double-buffer staged into LDS
//    with GLOBAL_LOAD_ASYNC_TO_LDS_B128 (ASYNCcnt-pipelined) and shared by
//    4 waves per block; argmin finished with xor-shuffles per 16-lane half.
// Workspace requirement: ~1.9 MB in d_ws.
// ---------------------------------------------------------------------------

#define BB        2
#define NN        16384
#define NPT       4096
#define HSZ       32768          // hash slots per batch (2x max uniques)
#define HMASK     (HSZ - 1)
#define EMPTYK    (-2147483647 - 1)
#define V0c       0.02f
#define KPc       0.5f
#define KIc       0.1f
#define MAXIT     10

// nearest-kernel tiling
#define WAVES     4
#define BTHREADS  (WAVES * 32)
#define ROWS      (WAVES * 16)        // centroid rows per block
#define CHUNK     1024                // points staged per buffer (12 KB)
#define NCHUNK    (NN / CHUNK)        // 16
#define OPS_PER_STAGE ((CHUNK * 12) / (BTHREADS * 16))   // 6 x b128 per thread

struct Ctrl {
    float scale;
    float serr;
    float vs;
    int   done;
    int   mmax;
    int   pad[11];
};

typedef __attribute__((ext_vector_type(2))) float v2f;
typedef __attribute__((ext_vector_type(8))) float v8f;

__device__ __forceinline__ int voxel_key(float x, float y, float z, float vs) {
    int c0 = (int)floorf(x / vs);
    int c1 = (int)floorf(y / vs);
    int c2 = (int)floorf(z / vs);
    return c0 * 1000000 + c1 * 1000 + c2;
}

// ---------------------------------------------------------------------------
__global__ void k_init_ctrl(Ctrl* ctrl) {
    if (threadIdx.x == 0 && blockIdx.x == 0) {
        ctrl->scale = 0.0f;
        ctrl->serr  = 0.0f;
        ctrl->vs    = V0c;
        ctrl->done  = 0;
        ctrl->mmax  = 1;
    }
}

__global__ void k_clear_hash(int* hashKeys, int* uniq) {
    int i = blockIdx.x * blockDim.x + threadIdx.x;
    if (i < BB * HSZ) hashKeys[i] = EMPTYK;
    if (i < BB)       uniq[i] = 0;
}

// one PI-loop iteration: insert every point's key, count fresh insertions
__global__ void k_count(const float* __restrict__ xyz, const Ctrl* __restrict__ ctrl,
                        int* __restrict__ hashKeys, int* __restrict__ uniq) {
    if (ctrl->done) return;
    int t = blockIdx.x * blockDim.x + threadIdx.x;
    if (t >= BB * NN) return;
    int b = t / NN;
    float vs = V0c * expf(ctrl->scale);
    const float* p = xyz + (size_t)t * 3;
    int key = voxel_key(p[0], p[1], p[2], vs);
    int* tab = hashKeys + b * HSZ;
    unsigned h = (unsigned)key * 2654435761u;
    int slot = (int)(h & HMASK);
    for (;;) {
        int prev = atomicCAS(&tab[slot], EMPTYK, key);
        if (prev == EMPTYK) { atomicAdd(&uniq[b], 1); break; }
        if (prev == key) break;
        slot = (slot + 1) & HMASK;
    }
}

__global__ void k_pi_update(Ctrl* ctrl, const int* __restrict__ uniq) {
    if (threadIdx.x != 0 || blockIdx.x != 0) return;
    if (ctrl->done) return;
    float scale = ctrl->scale;
    float vs = V0c * expf(scale);         // voxel size used THIS iteration
    int m = 1;
    for (int b = 0; b < BB; ++b) m = (uniq[b] > m) ? uniq[b] : m;
    const float target_ratio = (float)NN / (float)NPT;
    float err  = target_ratio - (float)NN / (float)m;
    float serr = ctrl->serr + err;
    float diff = KPc * err + KIc * serr;
    float sig  = 1.0f / (1.0f + expf(-diff));
    ctrl->vs    = vs;
    ctrl->serr  = serr;
    ctrl->scale = scale + 0.01f * (sig - 0.5f);
    float ad = fabsf((float)(m - NPT));
    if (ad <= (float)NPT * 0.1f) ctrl->done = 1;
}

// ---------------------------------------------------------------------------
__global__ void k_clear_build(int* hashKeys, float* sums, int* cnt,
                              int* ukeys, float* cent, int* Mcnt) {
    int i = blockIdx.x * blockDim.x + threadIdx.x;
    if (i < BB * HSZ) {
        hashKeys[i] = EMPTYK;
        cnt[i] = 0;
        sums[i * 3 + 0] = 0.0f;
        sums[i * 3 + 1] = 0.0f;
        sums[i * 3 + 2] = 0.0f;
    }
    if (i < BB * NN) {
        ukeys[i] = 2147483647;           // pad for ascending bitonic sort
        cent[i * 3 + 0] = 0.0f;          // zero padding rows (rank >= M)
        cent[i * 3 + 1] = 0.0f;
        cent[i * 3 + 2] = 0.0f;
    }
    if (i < BB) Mcnt[i] = 0;
}

__global__ void k_build(const float* __restrict__ xyz, const Ctrl* __restrict__ ctrl,
                        int* __restrict__ hashKeys, float* __restrict__ sums,
                        int* __restrict__ cnt, int* __restrict__ ukeys,
                        int* __restrict__ Mcnt) {
    int t = blockIdx.x * blockDim.x + threadIdx.x;
    if (t >= BB * NN) return;
    int b = t / NN;
    float vs = ctrl->vs;
    const float* p = xyz + (size_t)t * 3;
    float px = p[0], py = p[1], pz = p[2];
    int key = voxel_key(px, py, pz, vs);
    int* tab = hashKeys + b * HSZ;
    unsigned h = (unsigned)key * 2654435761u;
    int slot = (int)(h & HMASK);
    for (;;) {
        int prev = atomicCAS(&tab[slot], EMPTYK, key);
        if (prev == EMPTYK) {
            int idx = atomicAdd(&Mcnt[b], 1);
            ukeys[b * NN + idx] = key;   // unsorted append; sorted next
            break;
        }
        if (prev == key) break;
        slot = (slot + 1) & HMASK;
    }
    int g = b * HSZ + slot;
    atomicAdd(&sums[g * 3 + 0], px);
    atomicAdd(&sums[g * 3 + 1], py);
    atomicAdd(&sums[g * 3 + 2], pz);
    atomicAdd(&cnt[g], 1);
}

// per-batch bitonic sort of NN ints in LDS (64 KB of the 320 KB/WGP)
__global__ void k_sort(int* __restrict__ ukeys) {
    __shared__ int s[NN];
    int* u = ukeys + blockIdx.x * NN;
    for (int i = threadIdx.x; i < NN; i += blockDim.x) s[i] = u[i];
    __syncthreads();
    for (int k = 2; k <= NN; k <<= 1) {
        for (int j = k >> 1; j > 0; j >>= 1) {
            for (int i = threadIdx.x; i < NN; i += blockDim.x) {
                int ixj = i ^ j;
                if (ixj > i) {
                    int a = s[i], c = s[ixj];
                    bool up = ((i & k) == 0);
                    if ((a > c) == up) { s[i] = c; s[ixj] = a; }
                }
            }
            __syncthreads();
        }
    }
    for (int i = threadIdx.x; i < NN; i += blockDim.x) u[i] = s[i];
}

// rank = position of slot key in sorted unique keys; centroid = mean
__global__ void k_centroid(const int* __restrict__ hashKeys, const float* __restrict__ sums,
                           const int* __restrict__ cnt, const int* __restrict__ ukeys,
                           const int* __restrict__ Mcnt, float* __restrict__ cent) {
    int i = blockIdx.x * blockDim.x + threadIdx.x;
    if (i >= BB * HSZ) return;
    int key = hashKeys[i];
    if (key == EMPTYK) return;
    int b = i / HSZ;
    int Mb = Mcnt[b];
    const int* u = ukeys + b * NN;
    int lo = 0, hi = Mb;
    while (lo < hi) {                    // lower_bound (key is present)
        int mid = (lo + hi) >> 1;
        if (u[mid] < key) lo = mid + 1; else hi = mid;
    }
    float c = (float)cnt[i];
    c = (c < 1.0f) ? 1.0f : c;
    float inv = 1.0f / c;
    float* dst = cent + ((size_t)b * NN + lo) * 3;
    dst[0] = sums[i * 3 + 0] * inv;
    dst[1] = sums[i * 3 + 1] * inv;
    dst[2] = sums[i * 3 + 2] * inv;
}

__global__ void k_mmax(Ctrl* ctrl, const int* __restrict__ Mcnt) {
    if (threadIdx.x != 0 || blockIdx.x != 0) return;
    int m = 1;
    for (int b = 0; b < BB; ++b) m = (Mcnt[b] > m) ? Mcnt[b] : m;
    ctrl->mmax = m;
}

// ---------------------------------------------------------------------------
// Async staging: one b128 per thread per sweep, ASYNCcnt-tracked.
// LDS destination address = low 32 bits of the shared pointer's flat address
// (LDS aperture keeps the wave-relative byte offset in addr[31:0]).
// ---------------------------------------------------------------------------
__device__ __forceinline__ void stage_chunk(const float* __restrict__ xb, int chunk,
                                            uint32_t ldsbase, int tid) {
    unsigned long long g = (unsigned long long)(uintptr_t)(xb + (size_t)chunk * CHUNK * 3)
                         + (unsigned long long)tid * 16ull;
    uint32_t l = ldsbase + (uint32_t)tid * 16u;
#pragma unroll
    for (int k = 0; k < OPS_PER_STAGE; ++k) {
        asm volatile("global_load_async_to_lds_b128 %0, %1, off"
                     :: "v"(l), "v"(g) : "memory");
        g += (unsigned long long)(BTHREADS * 16);
        l += (uint32_t)(BTHREADS * 16);
    }
}

// ---------------------------------------------------------------------------
// WMMA nearest-point kernel. 4 waves per block; each wave owns a 16-row
// centroid tile; the block shares double-buffered LDS point chunks.
// score = x2[n] - 2*(x.c)  (c2[m] is constant per row -> drops from argmin).
// D layout: VGPR r, lane l -> M = r + 8*(l/16), N = l%16.
// ---------------------------------------------------------------------------
__global__ void __launch_bounds__(BTHREADS)
k_nearest(const float* __restrict__ xyz, const float* __restrict__ cent,
          int* __restrict__ nearest) {
    __shared__ __align__(16) float pbuf[2][CHUNK * 3];

    int tid  = (int)threadIdx.x;
    int wave = tid >> 5;
    int lane = tid & 31;
    int col  = lane & 15;
    bool hic = lane >= 16;

    int b  = blockIdx.x / (NPT / ROWS);
    int m0 = (blockIdx.x % (NPT / ROWS)) * ROWS + wave * 16;

    const float* xb = xyz + (size_t)b * NN * 3;

    // A fragment (centroids 16x4): lanes 0-15 hold K=0,1; lanes 16-31 K=2,3
    const float* cp = cent + ((size_t)b * NN + (m0 + col)) * 3;
    float cx = cp[0], cy = cp[1], cz = cp[2];
    v2f afrag;
    afrag.x = hic ? cz : cx;
    afrag.y = hic ? 0.0f : cy;

    float best[8];
    int   bidx[8];
#pragma unroll
    for (int r = 0; r < 8; ++r) { best[r] = 3.402823e38f; bidx[r] = 0; }

    uint32_t lds0 = (uint32_t)(uintptr_t)&pbuf[0][0];
    uint32_t lds1 = (uint32_t)(uintptr_t)&pbuf[1][0];

    stage_chunk(xb, 0, lds0, tid);                       // prologue: chunk 0
    for (int c = 0; c < NCHUNK; ++c) {
        if (c + 1 < NCHUNK) {
            stage_chunk(xb, c + 1, ((c + 1) & 1) ? lds1 : lds0, tid);
            // 6 ops just issued may stay in flight; chunk c's 6 must be done
            asm volatile("s_wait_asynccnt 0x6" ::: "memory");
        } else {
            asm volatile("s_wait_asynccnt 0x0" ::: "memory");
        }
        __syncthreads();                                 // chunk c visible to all waves

        const float* lp = &pbuf[c & 1][0];
        int base = c * CHUNK;
#pragma unroll 2
        for (int n = 0; n < CHUNK; n += 16) {
            const float* pp = lp + (size_t)(n + col) * 3;
            float px = pp[0], py = pp[1], pz = pp[2];
            float x2 = px * px + py * py + pz * pz;

            // B fragment (points 4x16): lanes 0-15 K=0,1; lanes 16-31 K=2,3
            v2f bfrag;
            bfrag.x = hic ? pz : px;
            bfrag.y = hic ? 0.0f : py;

            v8f acc = {0.0f, 0.0f, 0.0f, 0.0f, 0.0f, 0.0f, 0.0f, 0.0f};
            acc = __builtin_amdgcn_wmma_f32_16x16x4_f32(
                /*neg_a=*/false, afrag, /*neg_b=*/false, bfrag,
                /*c_mod=*/(short)0, acc, /*reuse_a=*/false, /*reuse_b=*/false);

#pragma unroll
            for (int r = 0; r < 8; ++r) {
                float s = x2 - 2.0f * acc[r];
                if (s < best[r]) { best[r] = s; bidx[r] = base + n + col; }
            }
        }
        __syncthreads();                                 // done reading buf before restage
    }

#pragma unroll
    for (int r = 0; r < 8; ++r) {
        float s = best[r];
        int   i = bidx[r];
#pragma unroll
        for (int off = 1; off < 16; off <<= 1) {         // stays within 16-lane half
            float s2 = __shfl_xor(s, off, 32);
            int   i2 = __shfl_xor(i, off, 32);
            if (s2 < s || (s2 == s && i2 < i)) { s = s2; i = i2; }
        }
        if (col == 0) {
            int m = m0 + r + (hic ? 8 : 0);
            nearest[b * NPT + m] = i;
        }
    }
}

__global__ void k_gather(const Ctrl* __restrict__ ctrl, const int* __restrict__ nearest,
                         int* __restrict__ out) {
    int t = blockIdx.x * blockDim.x + threadIdx.x;
    if (t >= BB * NPT) return;
    int b = t / NPT, j = t % NPT;
    int pos = ctrl->mmax - 1;
    if (j < pos) pos = j;
    out[t] = nearest[b * NPT + pos];
}

// ---------------------------------------------------------------------------
extern "C" void kernel_launch(void* const* d_in, const int* in_sizes, int n_in,
                              void* d_out, int out_size, void* d_ws, size_t ws_size,
                              hipStream_t stream) {
    const float* xyz = (const float*)d_in[0];
    int* out = (int*)d_out;                   // reference output dtype: int32

    // ---- workspace carve-up (all offsets 256B aligned) ----
    char* ws = (char*)d_ws;
    size_t off = 0;
    auto take = [&](size_t bytes) { void* p = ws + off; off = (off + bytes + 255) & ~(size_t)255; return p; };
    Ctrl*  ctrl     = (Ctrl*) take(sizeof(Ctrl));
    int*   hashKeys = (int*)  take((size_t)BB * HSZ * sizeof(int));
    int*   uniq     = (int*)  take((size_t)BB * sizeof(int));
    float* sums     = (float*)take((size_t)BB * HSZ * 3 * sizeof(float));
    int*   cnt      = (int*)  take((size_t)BB * HSZ * sizeof(int));
    int*   ukeys    = (int*)  take((size_t)BB * NN * sizeof(int));
    int*   Mcnt     = (int*)  take((size_t)BB * sizeof(int));
    float* cent     = (float*)take((size_t)BB * NN * 3 * sizeof(float));
    int*   nearest  = (int*)  take((size_t)BB * NPT * sizeof(int));
    (void)ws_size; (void)in_sizes; (void)n_in; (void)out_size;

    const int T = 256;
    const int gHash  = (BB * HSZ + T - 1) / T;
    const int gPts   = (BB * NN  + T - 1) / T;
    const int gOut   = (BB * NPT + T - 1) / T;

    k_init_ctrl<<<1, 32, 0, stream>>>(ctrl);

    // PI control loop: fixed MAXIT iterations, device "done" flag gates work
    for (int it = 0; it < MAXIT; ++it) {
        k_clear_hash<<<gHash, T, 0, stream>>>(hashKeys, uniq);
        k_count<<<gPts, T, 0, stream>>>(xyz, ctrl, hashKeys, uniq);
        k_pi_update<<<1, 32, 0, stream>>>(ctrl, uniq);
    }

    // centroid build with final voxel size
    k_clear_build<<<gHash, T, 0, stream>>>(hashKeys, sums, cnt, ukeys, cent, Mcnt);
    k_build<<<gPts, T, 0, stream>>>(xyz, ctrl, hashKeys, sums, cnt, ukeys, Mcnt);
    k_sort<<<BB, 1024, 0, stream>>>(ukeys);
    k_centroid<<<gHash, T, 0, stream>>>(hashKeys, sums, cnt, ukeys, Mcnt, cent);
    k_mmax<<<1, 32, 0, stream>>>(ctrl, Mcnt);

    // WMMA argmin: 4 waves per block sharing async-staged LDS point chunks
    k_nearest<<<BB * (NPT / ROWS), BTHREADS, 0, stream>>>(xyz, cent, nearest);
    k_gather<<<gOut, T, 0, stream>>>(ctrl, nearest, out);
}